// RendererModel_63788854280744
// MI455X (gfx1250) — compile-verified
//
#include <hip/hip_runtime.h>
#include <math.h>

typedef __attribute__((ext_vector_type(2))) float v2f;
typedef __attribute__((ext_vector_type(8))) float v8f;

#define SRES   128
#define NPIX   (SRES * SRES)
#define NFACES 2000
#define NTILES (NFACES / 16)   // 125
#define NF     NFACES
#define EPSV   1e-9f
#define DISTV  2.732f
#define TEXW   512
#define TEXH   512

// ---------------------------------------------------------------------------
// Kernel 1: per-face affine coefficients for w1, w2, depth as functions of
// (px, py). ws layout: A1 B1 C1 A2 B2 C2 Ad Bd Cd, each NFACES floats.
// ---------------------------------------------------------------------------
__global__ void k_setup(const float* __restrict__ verts,
                        const int* __restrict__ faces,
                        const int* __restrict__ azi,
                        const int* __restrict__ ele,
                        float* __restrict__ ws) {
  int f = blockIdx.x * blockDim.x + threadIdx.x;
  if (f >= NFACES) return;

  const float D2R = 3.14159265358979323846f / 180.0f;
  float a = (float)azi[0] * D2R;
  float e = (float)ele[0] * D2R;
  float ex = DISTV * (__cosf(e) * __sinf(a));
  float ey = DISTV * __sinf(e);
  float ez = -DISTV * (__cosf(e) * __cosf(a));

  float en = sqrtf(ex * ex + ey * ey + ez * ez);
  float zx = -ex / en, zy = -ey / en, zz = -ez / en;
  // x = cross(up=(0,1,0), z) = (zz, 0, -zx)
  float xx = zz, xy = 0.0f, xz = -zx;
  float xn = sqrtf(xx * xx + xy * xy + xz * xz);
  xx /= xn; xy /= xn; xz /= xn;
  // y = cross(z, x)
  float yx = zy * xz - zz * xy;
  float yy = zz * xx - zx * xz;
  float yz = zx * xy - zy * xx;
  float yn = sqrtf(yx * yx + yy * yy + yz * yz);
  yx /= yn; yy /= yn; yz /= yn;

  float P[3][3];
  #pragma unroll
  for (int k = 0; k < 3; k++) {
    int vi = faces[f * 3 + k];
    float vx = verts[vi * 3 + 0] - ex;
    float vy = verts[vi * 3 + 1] - ey;
    float vz = verts[vi * 3 + 2] - ez;
    P[k][0] = xx * vx + xy * vy + xz * vz;
    P[k][1] = yx * vx + yy * vy + yz * vz;
    P[k][2] = zx * vx + zy * vy + zz * vz;
  }

  float e1x = P[1][0] - P[0][0], e1y = P[1][1] - P[0][1];
  float e2x = P[2][0] - P[0][0], e2y = P[2][1] - P[0][1];
  float det = e1x * e2y - e1y * e2x;
  bool ok = fabsf(det) > EPSV;
  float inv = 1.0f / (ok ? det : 1.0f);

  float A1 = inv * e2y;
  float B1 = -inv * e2x;
  float C1 = inv * (P[0][1] * e2x - P[0][0] * e2y);
  float A2 = -inv * e1y;
  float B2 = inv * e1x;
  float C2 = inv * (e1y * P[0][0] - e1x * P[0][1]);
  if (!ok) {  // degenerate face: force w1,w2 < 0 forever -> never inside
    A1 = 0.0f; B1 = 0.0f; C1 = -1e30f;
    A2 = 0.0f; B2 = 0.0f; C2 = -1e30f;
  }
  float z0 = P[0][2];
  float dz1 = P[1][2] - z0, dz2 = P[2][2] - z0;
  float Ad = A1 * dz1 + A2 * dz2;
  float Bd = B1 * dz1 + B2 * dz2;
  float Cd = z0 + C1 * dz1 + C2 * dz2;

  ws[0 * NF + f] = A1; ws[1 * NF + f] = B1; ws[2 * NF + f] = C1;
  ws[3 * NF + f] = A2; ws[4 * NF + f] = B2; ws[5 * NF + f] = C2;
  ws[6 * NF + f] = Ad; ws[7 * NF + f] = Bd; ws[8 * NF + f] = Cd;
}

// ---------------------------------------------------------------------------
// Kernel 2: WMMA pixel x face sweep. Wave handles 16 pixels (one row chunk),
// loops over 125 face tiles; 3x v_wmma_f32_16x16x4_f32 per tile computes
// w1, w2, depth for the 16x16 (pixel, face) block. Running (depth,fid) min
// per element, then shfl_xor reduction over the 16 face columns.
// ---------------------------------------------------------------------------
__global__ __launch_bounds__(256) void k_raster(const float* __restrict__ ws,
                                                float* __restrict__ bestD,
                                                int* __restrict__ bestF) {
  int lane = threadIdx.x & 31;
  int wave = (blockIdx.x * blockDim.x + threadIdx.x) >> 5;  // 0..1023
  int pixBase = wave * 16;
  int y  = pixBase >> 7;   // row (16 pixels share a row: 128 % 16 == 0)
  int x0 = pixBase & 127;
  int m  = lane & 15;
  bool hi = lane >= 16;

  // A matrix 16x4: rows = pixels, K = (px, py, 1, 0).
  // lanes 0-15: K=0,1 ; lanes 16-31: K=2,3 (ISA 32-bit A layout).
  float px = ((float)(x0 + m) + 0.5f) * (2.0f / SRES) - 1.0f;
  float py = 1.0f - ((float)y + 0.5f) * (2.0f / SRES);
  v2f a;
  a.x = hi ? 1.0f : px;
  a.y = hi ? 0.0f : py;

  float bd_[8];
  int   bf_[8];
  #pragma unroll
  for (int i = 0; i < 8; i++) { bd_[i] = __builtin_inff(); bf_[i] = 0; }

  for (int t = 0; t < NTILES; t++) {
    int f = t * 16 + m;   // this lane's face column
    // B matrices 4x16: rows K = (A,B,C,0) coeffs, cols = faces.
    // lanes 0-15: K=0,1 ; lanes 16-31: K=2,3. Branchless (EXEC all-1 for WMMA).
    float s1a = ws[0 * NF + f], s1b = ws[1 * NF + f], s1c = ws[2 * NF + f];
    float s2a = ws[3 * NF + f], s2b = ws[4 * NF + f], s2c = ws[5 * NF + f];
    float sda = ws[6 * NF + f], sdb = ws[7 * NF + f], sdc = ws[8 * NF + f];
    v2f b1, b2, bd;
    b1.x = hi ? s1c : s1a;  b1.y = hi ? 0.0f : s1b;
    b2.x = hi ? s2c : s2a;  b2.y = hi ? 0.0f : s2b;
    bd.x = hi ? sdc : sda;  bd.y = hi ? 0.0f : sdb;

    v8f zero = {};
    v8f W1 = __builtin_amdgcn_wmma_f32_16x16x4_f32(
        false, a, false, b1, (short)0, zero, false, false);
    v8f W2 = __builtin_amdgcn_wmma_f32_16x16x4_f32(
        false, a, false, b2, (short)0, zero, false, false);
    v8f DP = __builtin_amdgcn_wmma_f32_16x16x4_f32(
        false, a, false, bd, (short)0, zero, false, false);

    #pragma unroll
    for (int i = 0; i < 8; i++) {
      float w1 = W1[i], w2 = W2[i];
      float w0 = 1.0f - w1 - w2;
      float dd = DP[i];
      bool inside = (w0 >= 0.0f) && (w1 >= 0.0f) && (w2 >= 0.0f);
      // faces visited in increasing fid order per element -> strict '<'
      // keeps the first (lowest-fid) minimum, matching jnp.argmin.
      if (inside && dd < bd_[i]) { bd_[i] = dd; bf_[i] = f; }
    }
  }

  // Reduce across the 16 face-columns (lane halves are independent pixels).
  #pragma unroll
  for (int i = 0; i < 8; i++) {
    float d = bd_[i];
    int   fi = bf_[i];
    #pragma unroll
    for (int off = 8; off > 0; off >>= 1) {
      float d2 = __shfl_xor(d, off, 32);
      int   f2 = __shfl_xor(fi, off, 32);
      if (d2 < d || (d2 == d && f2 < fi)) { d = d2; fi = f2; }
    }
    if (m == 0) {
      int pix = pixBase + (hi ? i + 8 : i);
      bestD[pix] = d;
      bestF[pix] = fi;
    }
  }
}

// ---------------------------------------------------------------------------
// Kernel 3: shade hit pixels. Recompute winning barycentrics, pick texture
// cell, bilinear grid-sample on demand, write x-flipped CHW output.
// ---------------------------------------------------------------------------
__global__ void k_shade(const float* __restrict__ ws,
                        const float* __restrict__ bestD,
                        const int* __restrict__ bestF,
                        const float* __restrict__ tex,
                        const float* __restrict__ grid,
                        float* __restrict__ out) {
  int p = blockIdx.x * blockDim.x + threadIdx.x;
  if (p >= NPIX) return;
  int y = p >> 7, x = p & 127;

  float c0 = 0.0f, c1 = 0.0f, c2 = 0.0f;
  float d = bestD[p];
  if (d < 3.0e38f) {  // finite -> hit
    int f = bestF[p];
    float px = ((float)x + 0.5f) * (2.0f / SRES) - 1.0f;
    float py = 1.0f - ((float)y + 0.5f) * (2.0f / SRES);
    float w1 = ws[0 * NF + f] * px + ws[1 * NF + f] * py + ws[2 * NF + f];
    float w2 = ws[3 * NF + f] * px + ws[4 * NF + f] * py + ws[5 * NF + f];
    float w0 = 1.0f - w1 - w2;
    int i0 = min(max((int)floorf(w0 * 4.0f), 0), 3);
    int i1 = min(max((int)floorf(w1 * 4.0f), 0), 3);
    int i2 = min(max((int)floorf(w2 * 4.0f), 0), 3);
    int g = f * 64 + i0 * 16 + i1 * 4 + i2;

    float gx = grid[2 * g + 0];
    float gy = grid[2 * g + 1];
    float sx = (gx + 1.0f) * (TEXW * 0.5f) - 0.5f;
    float sy = (gy + 1.0f) * (TEXH * 0.5f) - 0.5f;
    float fx0 = floorf(sx), fy0 = floorf(sy);
    float tx = sx - fx0, ty = sy - fy0;
    int ix0 = (int)fx0, iy0 = (int)fy0;

    float wts[4] = { (1.0f - tx) * (1.0f - ty), tx * (1.0f - ty),
                     (1.0f - tx) * ty,          tx * ty };
    int xs_[4] = { ix0, ix0 + 1, ix0,     ix0 + 1 };
    int ys_[4] = { iy0, iy0,     iy0 + 1, iy0 + 1 };
    #pragma unroll
    for (int t = 0; t < 4; t++) {
      int xi = xs_[t], yi = ys_[t];
      bool valid = (xi >= 0) && (xi < TEXW) && (yi >= 0) && (yi < TEXH);
      int xc = min(max(xi, 0), TEXW - 1);
      int yc = min(max(yi, 0), TEXH - 1);
      float w = valid ? wts[t] : 0.0f;
      int base = yc * TEXW + xc;
      c0 += tex[0 * TEXH * TEXW + base] * w;
      c1 += tex[1 * TEXH * TEXW + base] * w;
      c2 += tex[2 * TEXH * TEXW + base] * w;
    }
  }

  // out[(c*S + y)*S + j] = color of pixel (y, S-1-j)  -> write at j = S-1-x
  int xo = (SRES - 1) - x;
  out[(0 * SRES + y) * SRES + xo] = c0;
  out[(1 * SRES + y) * SRES + xo] = c1;
  out[(2 * SRES + y) * SRES + xo] = c2;
}

// ---------------------------------------------------------------------------
extern "C" void kernel_launch(void* const* d_in, const int* in_sizes, int n_in,
                              void* d_out, int out_size, void* d_ws, size_t ws_size,
                              hipStream_t stream) {
  const float* verts = (const float*)d_in[0];  // (1200,3)
  const float* tex   = (const float*)d_in[1];  // (1,3,512,512)
  const float* grid  = (const float*)d_in[2];  // (1,128000,2)
  const int*   faces = (const int*)d_in[3];    // (2000,3)
  const int*   azi   = (const int*)d_in[4];    // scalar
  const int*   ele   = (const int*)d_in[5];    // scalar
  float*       out   = (float*)d_out;          // (1,3,128,128)

  float* ws    = (float*)d_ws;
  float* coeff = ws;                       // 9 * NFACES floats
  float* bestD = ws + 9 * NFACES;          // NPIX floats
  int*   bestF = (int*)(ws + 9 * NFACES + NPIX);  // NPIX ints

  k_setup<<<(NFACES + 255) / 256, 256, 0, stream>>>(verts, faces, azi, ele, coeff);
  // 1024 waves of 16 pixels, 8 waves/block -> 128 blocks
  k_raster<<<128, 256, 0, stream>>>(coeff, bestD, bestF);
  k_shade<<<(NPIX + 255) / 256, 256, 0, stream>>>(coeff, bestD, bestF, tex, grid, out);
}